// CustomHyperSemanticMessagePassing_28870770163848
// MI455X (gfx1250) — compile-verified
//
#include <hip/hip_runtime.h>
#include <hip/hip_bf16.h>
#include <stdint.h>

#define N_NODES 4096
#define N_EDGES 4096
#define DM      256
#define EDIM    64
#define HEADS   8
#define DH      32
#define DEG     8
#define CARD    8
#define LTOT    64   // DEG*CARD

typedef __attribute__((ext_vector_type(2))) float v2f;
typedef __attribute__((ext_vector_type(8))) float v8f;

#if __has_builtin(__builtin_amdgcn_tensor_load_to_lds)
#define USE_TDM 1
#else
#define USE_TDM 0
#endif

#if USE_TDM
typedef unsigned int v4u __attribute__((ext_vector_type(4)));
typedef int          v8i __attribute__((ext_vector_type(8)));
typedef int          v4i __attribute__((ext_vector_type(4)));

// Issue one TDM 2D tile load: (tile_w x tile_h) f32 tile from a row-major
// tensor with row length `rowlen` (elements), `nrows` rows, row stride
// `stride` (elements), into LDS at byte offset lds_off (packed contiguous).
// D# layout per CDNA5 ISA.
// This toolchain exposes the 6-argument builtin:
//   (uint32x4 g0, int32x8 g1, int32x4 g2, int32x4 g3, int32x8 pad, i32 cpol)
__device__ __forceinline__ void tdm_load_tile_f32(const float* gbase,
                                                  unsigned lds_off,
                                                  int tile_w, int tile_h,
                                                  int rowlen, int nrows,
                                                  int stride) {
    unsigned long long ga = (unsigned long long)(uintptr_t)gbase;
    v4u g0;
    g0[0] = 1u;                                   // count=1, user descriptor
    g0[1] = lds_off;                              // lds_addr (bytes)
    g0[2] = (unsigned)(ga & 0xFFFFFFFFu);         // global_addr[31:0]
    g0[3] = (unsigned)((ga >> 32) & 0x01FFFFFFu)  // global_addr[56:32]
            | 0x80000000u;                        // type=2 ("image")
    v8i g1;
    g1[0] = (int)(2u << 16);                              // data_size=4B, mask=0
    g1[1] = (int)((unsigned)(rowlen & 0xFFFF) << 16);     // tensor_dim0[15:0]
    g1[2] = (int)(((unsigned)(rowlen >> 16) & 0xFFFFu) |
                  ((unsigned)(nrows & 0xFFFF) << 16));    // dim0 hi | dim1 lo
    g1[3] = (int)(((unsigned)(nrows >> 16) & 0xFFFFu) |
                  ((unsigned)(tile_w & 0xFFFF) << 16));   // dim1 hi | tile_dim0
    g1[4] = (int)((unsigned)(tile_h & 0xFFFF));           // tile_dim1 (tile_dim2=0)
    g1[5] = (int)(unsigned)stride;                        // tensor_dim0_stride lo
    g1[6] = 0;                                            // stride hi / dim1_stride
    g1[7] = 0;
    v4i z4 = {};
    v8i z8 = {};
    __builtin_amdgcn_tensor_load_to_lds(g0, g1, z4, z4, z8, 0);
}
#endif

// ---------------------------------------------------------------------------
// Kernel 1: one wave32 per incidence row -> node_of_edge (ascending col idx).
// Single coalesced 67MB pass; also zeroes the per-node scatter counters.
// ---------------------------------------------------------------------------
__global__ void k_row_indices(const float* __restrict__ inc,
                              int* __restrict__ node_of_edge,
                              int* __restrict__ counters) {
    int gid  = blockIdx.x * blockDim.x + threadIdx.x;
    if (gid < N_NODES) counters[gid] = 0;

    int lane = threadIdx.x & 31;
    int row  = gid >> 5;                 // one wave per edge row
    if (row >= N_EDGES) return;

    const float* r = inc + (size_t)row * N_NODES;
    int cnt = 0;
    for (int c0 = 0; c0 < N_NODES; c0 += 32) {
        float v = r[c0 + lane];
        unsigned m = __builtin_amdgcn_ballot_w32(v != 0.0f);
        if (v != 0.0f) {
            int pos = cnt + __popc(m & ((1u << lane) - 1u));
            if (pos < CARD) node_of_edge[row * CARD + pos] = c0 + lane;
        }
        cnt += __popc(m);
    }
}

// ---------------------------------------------------------------------------
// Kernel 2: scatter edge ids to member nodes (order fixed by sort kernel).
// ---------------------------------------------------------------------------
__global__ void k_scatter_edges(const int* __restrict__ node_of_edge,
                                int* __restrict__ counters,
                                int* __restrict__ tmp_edges) {
    int gid = blockIdx.x * blockDim.x + threadIdx.x;
    if (gid >= N_EDGES * CARD) return;
    int e = gid >> 3;
    int u = node_of_edge[gid];
    int pos = atomicAdd(&counters[u], 1);
    if (pos < DEG) tmp_edges[u * DEG + pos] = e;
}

// ---------------------------------------------------------------------------
// Kernel 3: sort each node's 8 edges ascending (matches top_k tie-break).
// ---------------------------------------------------------------------------
__global__ void k_sort_edges(const int* __restrict__ tmp_edges,
                             int* __restrict__ edge_of_node) {
    int n = blockIdx.x * blockDim.x + threadIdx.x;
    if (n >= N_NODES) return;
    int v[DEG];
#pragma unroll
    for (int i = 0; i < DEG; ++i) v[i] = tmp_edges[n * DEG + i];
#pragma unroll
    for (int i = 1; i < DEG; ++i) {
        int key = v[i], j = i - 1;
        while (j >= 0 && v[j] > key) { v[j + 1] = v[j]; --j; }
        v[j + 1] = key;
    }
#pragma unroll
    for (int i = 0; i < DEG; ++i) edge_of_node[n * DEG + i] = v[i];
}

// ---------------------------------------------------------------------------
// WMMA FP32 GEMM. Block = 8 waves = 8 M-tiles sharing one 16-col N-tile.
// The B tile (16 x K) is staged once per block into LDS by the Tensor Data
// Mover (s_wait_tensorcnt fence), then WMMA B fragments come from ds loads.
// TB=false: B is (K,N) row-major.  TB=true: C = A * B^T with B (N,K) row-major.
// Fragment layouts per ISA: A 16x4 (lanes 0-15 rows, v0/v1 = K+2hi,K+2hi+1),
// B 4x16 (v0 = row k+2hi, v1 = row k+2hi+1), C VGPR v -> row m0+8hi+v.
// ---------------------------------------------------------------------------
template <bool TB, bool RELU>
__global__ void k_gemm_wmma(const float* __restrict__ A,
                            const float* __restrict__ B,
                            const float* __restrict__ bias,
                            float* __restrict__ C,
                            int M, int K, int N) {
    __shared__ float lds_b[16 * 256];        // max K = 256 -> 16KB

    const int t    = threadIdx.x;
    const int lane = t & 31;
    const int wib  = t >> 5;
    const int ntiles = N >> 4;
    const int nb = blockIdx.x % ntiles;
    const int mb = blockIdx.x / ntiles;
    const int n0 = nb * 16;
    const int m0 = mb * 128 + wib * 16;

    // ---- stage B tile: TB -> [16 rows][K]  |  !TB -> [K rows][16] ----
#if USE_TDM
    if (wib == 0) {
        unsigned lds_off = (unsigned)(uintptr_t)(void*)lds_b;
        if (TB) {
            // 16 rows (n0..n0+15) x K contiguous elements, row stride K
            tdm_load_tile_f32(B + (size_t)n0 * K, lds_off, K, 16, K, N, K);
        } else {
            // K rows x 16 elements starting at column n0, row stride N
            tdm_load_tile_f32(B + n0, lds_off, 16, K, N, K, N);
        }
        __builtin_amdgcn_s_wait_tensorcnt(0);
    }
#else
    {
        const int total = 16 * K;
        for (int i = t; i < total; i += 256) {
            if (TB) lds_b[i] = B[(size_t)(n0 + i / K) * K + (i % K)];
            else    lds_b[i] = B[(size_t)(i / 16) * N + n0 + (i % 16)];
        }
    }
#endif
    __syncthreads();
    if (m0 >= M) return;

    const int lr = lane & 15;     // row (A) / col (B,C) within tile
    const int hi = lane >> 4;     // K sub-group select

    v8f c = {};
    const float* arow = A + (size_t)(m0 + lr) * K + 2 * hi;
    const float* brow = &lds_b[lr * K + 2 * hi];     // used when TB
    for (int k = 0; k < K; k += 4) {
        v2f a = *(const v2f*)(arow + k);
        v2f b;
        if (TB) {
            b = *(const v2f*)(brow + k);
        } else {
            b.x = lds_b[(k + 2 * hi) * 16 + lr];
            b.y = lds_b[(k + 2 * hi + 1) * 16 + lr];
        }
        c = __builtin_amdgcn_wmma_f32_16x16x4_f32(
                /*neg_a=*/false, a, /*neg_b=*/false, b,
                /*c_mod=*/(short)0, c, /*reuse_a=*/false, /*reuse_b=*/false);
    }

    const int crow = m0 + hi * 8;
    const int ccol = n0 + lr;
    const float bb = bias ? bias[ccol] : 0.0f;
#pragma unroll
    for (int v = 0; v < 8; ++v) {
        float val = c[v] + bb;
        if (RELU) val = fmaxf(val, 0.0f);
        C[(size_t)(crow + v) * N + ccol] = val;
    }
}

// ---------------------------------------------------------------------------
// Attention: 1 block per node, 1 wave per head. Each lane owns two of the 64
// (edge,member) slots and accumulates its scores with b128 loads (no shuffles
// in the inner loop); softmax needs only two 5-step lane reductions; the V
// pass reads weights from LDS and gathers Vh coalesced (L2-resident).
// ---------------------------------------------------------------------------
__global__ void k_attention(const float* __restrict__ Q,
                            const float* __restrict__ Kh,
                            const float* __restrict__ Ke,
                            const float* __restrict__ Vh,
                            const int* __restrict__ edge_of_node,
                            const int* __restrict__ node_of_edge,
                            float* __restrict__ ctx) {
    const int n = blockIdx.x;
    __shared__ int   s_u[LTOT];
    __shared__ int   s_e[LTOT];
    __shared__ float s_q[DM];
    __shared__ float s_w[HEADS][LTOT];

    const int t = threadIdx.x;
    s_q[t] = Q[(size_t)n * DM + t];
    if (t < LTOT) {
        int e = edge_of_node[n * DEG + (t >> 3)];
        s_e[t] = e;
        s_u[t] = node_of_edge[e * CARD + (t & 7)];
    }
    __syncthreads();

    const int h    = t >> 5;
    const int lane = t & 31;

    // lane owns slots lA = lane, lB = lane + 32
    const int uA = s_u[lane],      eA = s_e[lane];
    const int uB = s_u[lane + 32], eB = s_e[lane + 32];
    const float* khA = Kh + (size_t)uA * DM + h * DH;
    const float* keA = Ke + (size_t)eA * DM + h * DH;
    const float* khB = Kh + (size_t)uB * DM + h * DH;
    const float* keB = Ke + (size_t)eB * DM + h * DH;
    const float* qh  = &s_q[h * DH];

    float sA = 0.0f, sB = 0.0f;
#pragma unroll
    for (int d = 0; d < DH; d += 4) {
        float4 qv = *(const float4*)(qh + d);
        float4 a1 = *(const float4*)(khA + d);
        float4 a2 = *(const float4*)(keA + d);
        float4 b1 = *(const float4*)(khB + d);
        float4 b2 = *(const float4*)(keB + d);
        sA += qv.x * (a1.x + a2.x) + qv.y * (a1.y + a2.y)
            + qv.z * (a1.z + a2.z) + qv.w * (a1.w + a2.w);
        sB += qv.x * (b1.x + b2.x) + qv.y * (b1.y + b2.y)
            + qv.z * (b1.z + b2.z) + qv.w * (b1.w + b2.w);
    }
    const float scale = 0.17677669529663689f;    // 1/sqrt(32)
    sA *= scale;
    sB *= scale;

    float m = fmaxf(sA, sB);
#pragma unroll
    for (int off = 16; off > 0; off >>= 1) m = fmaxf(m, __shfl_xor(m, off, 32));
    const float e0 = __expf(sA - m);
    const float e1 = __expf(sB - m);
    float sum = e0 + e1;
#pragma unroll
    for (int off = 16; off > 0; off >>= 1) sum += __shfl_xor(sum, off, 32);
    const float inv = 1.0f / sum;
    s_w[h][lane]      = e0 * inv;    // read back only by this wave: DScnt
    s_w[h][lane + 32] = e1 * inv;    // ordering suffices, no barrier needed

    float acc = 0.0f;
    for (int l = 0; l < LTOT; ++l) {
        acc += s_w[h][l] * Vh[(size_t)s_u[l] * DM + h * DH + lane];
    }
    ctx[(size_t)n * DM + h * DH + lane] = acc;
}

// ---------------------------------------------------------------------------
extern "C" void kernel_launch(void* const* d_in, const int* in_sizes, int n_in,
                              void* d_out, int out_size, void* d_ws, size_t ws_size,
                              hipStream_t stream) {
    const float* x          = (const float*)d_in[0];
    const float* incidence  = (const float*)d_in[1];
    const float* edge_attr  = (const float*)d_in[2];
    const float* W_lin      = (const float*)d_in[3];
    const float* W_edge     = (const float*)d_in[4];
    const float* in_proj_w  = (const float*)d_in[5];   // (768,256): Wq|Wk|Wv
    const float* in_proj_b  = (const float*)d_in[6];   // (768,)
    const float* out_proj_w = (const float*)d_in[7];   // (256,256)
    const float* out_proj_b = (const float*)d_in[8];
    float* out = (float*)d_out;

    // workspace carve-up (~28.5 MB)
    const size_t MAT = (size_t)N_NODES * DM;
    float* Wh  = (float*)d_ws;
    float* We  = Wh  + MAT;
    float* Qm  = We  + MAT;
    float* Kh  = Qm  + MAT;
    float* Vh  = Kh  + MAT;
    float* KeM = Vh  + MAT;
    float* ctx = KeM + MAT;
    int* node_of_edge = (int*)(ctx + MAT);
    int* edge_of_node = node_of_edge + N_EDGES * CARD;
    int* tmp_edges    = edge_of_node + N_NODES * DEG;
    int* counters     = tmp_edges    + N_NODES * DEG;

    // 1) topology extraction
    k_row_indices<<<(N_EDGES * 32) / 256, 256, 0, stream>>>(incidence, node_of_edge, counters);
    k_scatter_edges<<<(N_EDGES * CARD + 255) / 256, 256, 0, stream>>>(node_of_edge, counters, tmp_edges);
    k_sort_edges<<<(N_NODES + 255) / 256, 256, 0, stream>>>(tmp_edges, edge_of_node);

    // 2) dense projections via FP32 WMMA; B tile staged once per block by TDM
    const int GB = (N_NODES / 128) * (DM / 16);        // 512 blocks
    // Wh = x @ W_lin
    k_gemm_wmma<false, false><<<GB, 256, 0, stream>>>(x, W_lin, nullptr, Wh, N_NODES, DM, DM);
    // We = edge_attr @ W_edge
    k_gemm_wmma<false, false><<<GB, 256, 0, stream>>>(edge_attr, W_edge, nullptr, We, N_EDGES, EDIM, DM);
    // Q = Wh @ Wq^T + bq
    k_gemm_wmma<true, false><<<GB, 256, 0, stream>>>(Wh, in_proj_w, in_proj_b, Qm, N_NODES, DM, DM);
    // Kh = Wh @ Wk^T              (bk folded into Ke)
    k_gemm_wmma<true, false><<<GB, 256, 0, stream>>>(Wh, in_proj_w + DM * DM, nullptr, Kh, N_NODES, DM, DM);
    // Vh = Wh @ Wv^T + bv
    k_gemm_wmma<true, false><<<GB, 256, 0, stream>>>(Wh, in_proj_w + 2 * DM * DM, in_proj_b + 2 * DM, Vh, N_NODES, DM, DM);
    // Ke = We @ Wk^T + bk
    k_gemm_wmma<true, false><<<GB, 256, 0, stream>>>(We, in_proj_w + DM * DM, in_proj_b + DM, KeM, N_EDGES, DM, DM);

    // 3) gather attention (all operands L2-resident)
    k_attention<<<N_NODES, 256, 0, stream>>>(Qm, Kh, KeM, Vh, edge_of_node, node_of_edge, ctx);

    // 4) out = relu(ctx @ out_proj_w^T + b)
    k_gemm_wmma<true, true><<<GB, 256, 0, stream>>>(ctx, out_proj_w, out_proj_b, out, N_NODES, DM, DM);
}